// EdgeConv_12429635354789
// MI455X (gfx1250) — compile-verified
//
#include <hip/hip_runtime.h>

// ---------------------------------------------------------------------------
// EdgeConv on MI455X (gfx1250, wave32).
// fp32 WMMA (V_WMMA_F32_16X16X4_F32) for both dense layers; scatter/gather
// via global atomics; O(E^2) reverse-edge mask with LDS-resident indices.
// Total work ~220 MFLOP / ~15 MB -> latency bound; fp32 kept end-to-end.
// ---------------------------------------------------------------------------

typedef float v2f __attribute__((ext_vector_type(2)));
typedef float v8f __attribute__((ext_vector_type(8)));

#define N_NODES 20000
#define N_EDGES 8192
#define D_NODE  64
#define D_EDGE  16
#define UNITS   64
#define K1      (D_NODE + D_EDGE)   // 80
#define K2      (UNITS + UNITS)     // 128

// -------------------- GEMM1: edge_state = [X1] @ W_init + b ----------------
// X1 row e = concat(node_feature[edge_src[e]], edge_feature[e])   (len 80)
// One wave computes a 16-edge x 64-unit tile: 20 K-steps x 4 N-tiles = 80 WMMA.
__global__ __launch_bounds__(256) void gemm_init_kernel(
    const float* __restrict__ node_feature,
    const float* __restrict__ edge_feature,
    const float* __restrict__ W,        // [80][64] row-major
    const float* __restrict__ bias,     // [64]
    const int*   __restrict__ edge_src,
    float*       __restrict__ edge_state)
{
    __shared__ float sW[K1 * UNITS];    // 20 KB staged weights
    for (int i = threadIdx.x; i < K1 * UNITS; i += 256) sW[i] = W[i];
    __syncthreads();

    const int wave    = threadIdx.x >> 5;
    const int lane    = threadIdx.x & 31;
    const int rowTile = blockIdx.x * 8 + wave;        // 512 tiles total
    const int m       = lane & 15;                    // row within tile / col within N-tile
    const int kh      = (lane >> 4) << 1;             // K sub-select: 0 or 2
    const int e       = rowTile * 16 + m;
    const int s       = edge_src[e];
    const float* xn = node_feature + (long)s * D_NODE;
    const float* xe = edge_feature + (long)e * D_EDGE;

    v8f acc[4] = {};                                  // 16x64 fp32 output tile
    #pragma unroll
    for (int k0 = 0; k0 < K1; k0 += 4) {
        const int k = k0 + kh;                        // pair {k,k+1} never straddles 64
        v2f a;
        a.x = (k     < D_NODE) ? xn[k]     : xe[k     - D_NODE];
        a.y = (k + 1 < D_NODE) ? xn[k + 1] : xe[k + 1 - D_NODE];
        const float* wr0 = &sW[(k    ) * UNITS];
        const float* wr1 = &sW[(k + 1) * UNITS];
        #pragma unroll
        for (int nt = 0; nt < 4; ++nt) {
            v2f b;
            b.x = wr0[nt * 16 + m];
            b.y = wr1[nt * 16 + m];
            acc[nt] = __builtin_amdgcn_wmma_f32_16x16x4_f32(
                false, a, false, b, (short)0, acc[nt], false, false);
        }
    }

    const int half = lane >> 4;
    #pragma unroll
    for (int nt = 0; nt < 4; ++nt) {
        const int n = nt * 16 + m;
        const float bv = bias[n];
        #pragma unroll
        for (int r = 0; r < 8; ++r) {
            const int mm = r + 8 * half;              // C/D VGPR layout
            edge_state[(long)(rowTile * 16 + mm) * UNITS + n] = acc[nt][r] + bv;
        }
    }
}

// -------------------- zero node_agg ---------------------------------------
__global__ void zero_kernel(float* __restrict__ p, int n)
{
    int i = blockIdx.x * blockDim.x + threadIdx.x;
    if (i < n) p[i] = 0.0f;
}

// -------------------- segment_sum scatter ---------------------------------
__global__ void scatter_kernel(const float* __restrict__ edge_state,
                               const int*   __restrict__ edge_dst,
                               float*       __restrict__ node_agg)
{
    int t = blockIdx.x * 256 + threadIdx.x;           // E*UNITS threads
    int e = t >> 6, u = t & 63;
    atomicAdd(&node_agg[(long)edge_dst[e] * UNITS + u], edge_state[t]);
}

// -------------------- gather message --------------------------------------
__global__ void gather_kernel(const float* __restrict__ node_agg,
                              const int*   __restrict__ edge_src,
                              float*       __restrict__ message)
{
    int t = blockIdx.x * 256 + threadIdx.x;
    int e = t >> 6, u = t & 63;
    message[t] = node_agg[(long)edge_src[e] * UNITS + u];
}

// -------------------- reverse-edge mask subtraction -----------------------
// message[i] -= sum_j [src[j]==dst[i] && dst[j]==src[i]] * edge_state[j]
// Indices staged in 64 KB LDS; matches are rare so the float loop is cold.
// Each edge i is owned by exactly one thread -> plain (non-atomic) updates.
__global__ __launch_bounds__(256) void reverse_kernel(
    const int*   __restrict__ edge_src,
    const int*   __restrict__ edge_dst,
    const float* __restrict__ edge_state,
    float*       __restrict__ message)
{
    __shared__ int ss[N_EDGES];
    __shared__ int sd[N_EDGES];
    for (int j = threadIdx.x; j < N_EDGES; j += 256) {
        ss[j] = edge_src[j];
        sd[j] = edge_dst[j];
    }
    __syncthreads();

    const int i  = blockIdx.x * 256 + threadIdx.x;
    const int si = ss[i];
    const int di = sd[i];
    float* mi = message + (long)i * UNITS;
    for (int j = 0; j < N_EDGES; ++j) {
        if (ss[j] == di && sd[j] == si) {
            const float* esj = edge_state + (long)j * UNITS;
            #pragma unroll 4
            for (int u = 0; u < UNITS; ++u) mi[u] -= esj[u];
        }
    }
}

// -------------------- GEMM2: out = [edge_state | message] @ W_upd + b -----
__global__ __launch_bounds__(256) void gemm_upd_kernel(
    const float* __restrict__ edge_state,
    const float* __restrict__ message,
    const float* __restrict__ W,        // [128][64] row-major
    const float* __restrict__ bias,     // [64]
    float*       __restrict__ out)
{
    __shared__ float sW[K2 * UNITS];    // 32 KB staged weights
    for (int i = threadIdx.x; i < K2 * UNITS; i += 256) sW[i] = W[i];
    __syncthreads();

    const int wave    = threadIdx.x >> 5;
    const int lane    = threadIdx.x & 31;
    const int rowTile = blockIdx.x * 8 + wave;
    const int m       = lane & 15;
    const int kh      = (lane >> 4) << 1;
    const int e       = rowTile * 16 + m;
    const float* x0 = edge_state + (long)e * UNITS;
    const float* x1 = message    + (long)e * UNITS;

    v8f acc[4] = {};
    #pragma unroll
    for (int k0 = 0; k0 < K2; k0 += 4) {
        const int k = k0 + kh;
        v2f a;
        a.x = (k     < UNITS) ? x0[k]     : x1[k     - UNITS];
        a.y = (k + 1 < UNITS) ? x0[k + 1] : x1[k + 1 - UNITS];
        const float* wr0 = &sW[(k    ) * UNITS];
        const float* wr1 = &sW[(k + 1) * UNITS];
        #pragma unroll
        for (int nt = 0; nt < 4; ++nt) {
            v2f b;
            b.x = wr0[nt * 16 + m];
            b.y = wr1[nt * 16 + m];
            acc[nt] = __builtin_amdgcn_wmma_f32_16x16x4_f32(
                false, a, false, b, (short)0, acc[nt], false, false);
        }
    }

    const int half = lane >> 4;
    #pragma unroll
    for (int nt = 0; nt < 4; ++nt) {
        const int n = nt * 16 + m;
        const float bv = bias[n];
        #pragma unroll
        for (int r = 0; r < 8; ++r) {
            const int mm = r + 8 * half;
            out[(long)(rowTile * 16 + mm) * UNITS + n] = acc[nt][r] + bv;
        }
    }
}

// ---------------------------------------------------------------------------
extern "C" void kernel_launch(void* const* d_in, const int* in_sizes, int n_in,
                              void* d_out, int out_size, void* d_ws, size_t ws_size,
                              hipStream_t stream)
{
    const float* node_feature = (const float*)d_in[0];
    const float* edge_feature = (const float*)d_in[1];
    const float* W_init       = (const float*)d_in[2];
    const float* b_init       = (const float*)d_in[3];
    const float* W_upd        = (const float*)d_in[4];
    const float* b_upd        = (const float*)d_in[5];
    const int*   edge_src     = (const int*)d_in[6];
    const int*   edge_dst     = (const int*)d_in[7];
    float*       out          = (float*)d_out;

    char* ws = (char*)d_ws;
    float* edge_state = (float*)ws;                                        // 2   MB
    float* node_agg   = (float*)(ws + (size_t)N_EDGES * UNITS * 4);        // 5.12MB
    float* message    = (float*)(ws + (size_t)N_EDGES * UNITS * 4
                                    + (size_t)N_NODES * UNITS * 4);        // 2   MB

    // 512 row-tiles of 16 edges, 8 waves (256 thr) per block -> 64 blocks
    gemm_init_kernel<<<64, 256, 0, stream>>>(node_feature, edge_feature,
                                             W_init, b_init, edge_src, edge_state);

    zero_kernel<<<(N_NODES * UNITS + 255) / 256, 256, 0, stream>>>(
        node_agg, N_NODES * UNITS);

    scatter_kernel<<<(N_EDGES * UNITS) / 256, 256, 0, stream>>>(
        edge_state, edge_dst, node_agg);

    gather_kernel<<<(N_EDGES * UNITS) / 256, 256, 0, stream>>>(
        node_agg, edge_src, message);

    reverse_kernel<<<N_EDGES / 256, 256, 0, stream>>>(
        edge_src, edge_dst, edge_state, message);

    gemm_upd_kernel<<<64, 256, 0, stream>>>(edge_state, message,
                                            W_upd, b_upd, out);
}